// DepthwiseConv3DTranspose_30837865185293
// MI455X (gfx1250) — compile-verified
//
#include <hip/hip_runtime.h>

// Depthwise 3D transposed conv, stride 2, k=4, SAME.
// x: (4,32,32,32,256) f32, filters: (4,4,4,1,256) f32, y: (4,64,64,64,256) f32.
//
// Parity decomposition (per spatial dim, od = 2q + r):
//   r=0: y += x[q-1]*F[3] + x[q]*F[1]
//   r=1: y += x[q]  *F[2] + x[q+1]*F[0]
// (F = original filters; the reference's kernel flip cancels.)
// => per output voxel: 2x2x2 = 8 taps.

typedef float f4 __attribute__((ext_vector_type(4)));
typedef __attribute__((address_space(3))) f4 lds_f4;

#define C4 64  // 256 channels as float4

__global__ __launch_bounds__(256)
void dwconv3dT_kernel(const float* __restrict__ xg,
                      const float* __restrict__ fg,
                      float* __restrict__ yg)
{
    __shared__ f4 sF[8][C4];   // 8 taps x 64 float4 = 8 KB

    const f4* x4  = (const f4*)xg;
    const f4* f4p = (const f4*)fg;
    f4*       y4  = (f4*)yg;

    const int ch  = threadIdx.x;                  // 0..63 channel-vec
    const int tid = threadIdx.y * 64 + threadIdx.x;

    // 8192 blocks total; 1024 blocks per parity -> parity uniform per block
    const int parity = blockIdx.x >> 10;
    const int rd = (parity >> 2) & 1;
    const int rh = (parity >> 1) & 1;
    const int rw =  parity       & 1;

    // ---- stage this parity's 8 filter taps into LDS via async DMA ----
    // tap t = a*4 + b*2 + c  (a: d-plane 0/1, b: h-plane, c: w-plane old/new)
    // kernel index per dim: k = 3 - r - 2*a
#pragma unroll
    for (int i = 0; i < 2; ++i) {
        const int e  = tid + i * 256;             // 512 float4s, 2 per thread
        const int t  = e >> 6;
        const int cc = e & 63;
        const int ka = 3 - rd - 2 * ((t >> 2) & 1);
        const int kb = 3 - rh - 2 * ((t >> 1) & 1);
        const int kc = 3 - rw - 2 * (t & 1);
        const f4* src = f4p + ((ka * 4 + kb) * 4 + kc) * C4 + cc;
        unsigned laddr = (unsigned)(unsigned long long)(lds_f4*)&sF[t][cc];
        unsigned long long gaddr = (unsigned long long)src;
        asm volatile("global_load_async_to_lds_b128 %0, %1, off"
                     :: "v"(laddr), "v"(gaddr) : "memory");
    }
    asm volatile("s_wait_asynccnt 0" ::: "memory");
    __syncthreads();

    f4 F[8];
#pragma unroll
    for (int t = 0; t < 8; ++t) F[t] = sF[t][ch];  // ds_load_b128 x8

    // ---- decode this thread's output row ----
    const int rid = (blockIdx.x << 2) + threadIdx.y;   // 0..32767
    const int rem = rid & 4095;
    const int n   = rem >> 10;
    const int qd  = (rem >> 5) & 31;
    const int qh  = rem & 31;

    const int zd0 = qd + rd - 1, zd1 = zd0 + 1;
    const int zh0 = qh + rh - 1, zh1 = zh0 + 1;
    const bool vd0 = (unsigned)zd0 < 32u, vd1 = (unsigned)zd1 < 32u;
    const bool vh0 = (unsigned)zh0 < 32u, vh1 = (unsigned)zh1 < 32u;
    const bool v00 = vd0 && vh0, v01 = vd0 && vh1;
    const bool v10 = vd1 && vh0, v11 = vd1 && vh1;

    const int zd0c = min(max(zd0, 0), 31), zd1c = min(max(zd1, 0), 31);
    const int zh0c = min(max(zh0, 0), 31), zh1c = min(max(zh1, 0), 31);

    const f4* b00 = x4 + (long)(((n * 32 + zd0c) * 32 + zh0c) * 32) * C4 + ch;
    const f4* b01 = x4 + (long)(((n * 32 + zd0c) * 32 + zh1c) * 32) * C4 + ch;
    const f4* b10 = x4 + (long)(((n * 32 + zd1c) * 32 + zh0c) * 32) * C4 + ch;
    const f4* b11 = x4 + (long)(((n * 32 + zd1c) * 32 + zh1c) * 32) * C4 + ch;

    const f4 zero = 0.0f;

    // initial "old" w-plane: zw = rw - 1 (valid only for rw==1, zw=0)
    f4 o00 = zero, o01 = zero, o10 = zero, o11 = zero;
    if (rw == 1) {
        o00 = v00 ? b00[0] : zero;
        o01 = v01 ? b01[0] : zero;
        o10 = v10 ? b10[0] : zero;
        o11 = v11 ? b11[0] : zero;
    }

    const int od = 2 * qd + rd, oh = 2 * qh + rh;
    f4* outp = y4 + (long)(((n * 64 + od) * 64 + oh) * 64 + rw) * C4 + ch;

#pragma unroll 4
    for (int qw = 0; qw < 32; ++qw) {
        const int  zw = qw + rw;                 // new w-plane
        const bool vw = zw < 32;
        const long off = (long)zw * C4;

        f4 n00 = (vw && v00) ? b00[off] : zero;
        f4 n01 = (vw && v01) ? b01[off] : zero;
        f4 n10 = (vw && v10) ? b10[off] : zero;
        f4 n11 = (vw && v11) ? b11[off] : zero;

        f4 acc;
        acc  = o00 * F[0];
        acc += n00 * F[1];
        acc += o01 * F[2];
        acc += n01 * F[3];
        acc += o10 * F[4];
        acc += n10 * F[5];
        acc += o11 * F[6];
        acc += n11 * F[7];

        // streaming output: keep it out of L2 so the input stays resident
        __builtin_nontemporal_store(acc, outp);
        outp += 2 * C4;

        o00 = n00; o01 = n01; o10 = n10; o11 = n11;

        if ((qw & 7) == 0) {                     // prefetch 8 planes ahead
            const int pz = zw + 8;
            if (pz < 32) {
                const long poff = (long)pz * C4;
                if (v00) __builtin_prefetch(b00 + poff, 0, 1);
                if (v01) __builtin_prefetch(b01 + poff, 0, 1);
                if (v10) __builtin_prefetch(b10 + poff, 0, 1);
                if (v11) __builtin_prefetch(b11 + poff, 0, 1);
            }
        }
    }
}

extern "C" void kernel_launch(void* const* d_in, const int* in_sizes, int n_in,
                              void* d_out, int out_size, void* d_ws, size_t ws_size,
                              hipStream_t stream) {
    const float* x = (const float*)d_in[0];   // 4*32*32*32*256
    const float* f = (const float*)d_in[1];   // 4*4*4*1*256
    float*       y = (float*)d_out;           // 4*64*64*64*256

    // rows = 8 parities * 4 batch * 32 qd * 32 qh = 32768; 4 rows/block
    dim3 block(64, 4, 1);
    dim3 grid(8192, 1, 1);
    hipLaunchKernelGGL(dwconv3dT_kernel, grid, block, 0, stream, x, f, y);
}